// SSA_8950711846130
// MI455X (gfx1250) — compile-verified
//
#include <hip/hip_runtime.h>
#include <hip/hip_bf16.h>
#include <math.h>

// Problem constants (from setup_inputs)
#define NB   16      // batch
#define CH   256     // channels
#define HH   96
#define WW_  96
#define PLANE (HH*WW_)      // 9216 floats per (n,c) plane
#define GROUP_ 8
#define KTAP 7
#define FDIM (GROUP_*KTAP)  // 56

typedef __attribute__((ext_vector_type(2))) float v2f;
typedef __attribute__((ext_vector_type(8))) float v8f;

// ---------------------------------------------------------------------------
// Async copy of one 36 KB plane (global -> LDS) via CDNA5 async-LDS path.
// Each of the 256 threads issues 9 x B128 (16 B) transfers; ASYNCcnt tracked.
// ---------------------------------------------------------------------------
__device__ __forceinline__ void async_plane_to_lds(const float* __restrict__ gsrc,
                                                   float* lds_tile)
{
    // Low 32 bits of a generic pointer to LDS == byte offset in LDS space.
    const unsigned int lds_base = (unsigned int)(unsigned long long)(void*)lds_tile;
    const unsigned long long gbase = (unsigned long long)gsrc;
    #pragma unroll
    for (int j = 0; j < (PLANE * 4) / (256 * 16); ++j) {   // 9 iterations
        const unsigned int boff = (unsigned int)(j * 256 + threadIdx.x) * 16u;
        const unsigned int la = lds_base + boff;
        const unsigned long long ga = gbase + boff;
        asm volatile("global_load_async_to_lds_b128 %0, %1, off"
                     :: "v"(la), "v"(ga) : "memory");
    }
    asm volatile("s_wait_asynccnt 0" ::: "memory");  // drain this wave's async ops
}

// ---------------------------------------------------------------------------
// Kernel 1: per-(n,c) plane sum (one block per plane)
// ---------------------------------------------------------------------------
__global__ __launch_bounds__(256) void plane_sum_kernel(
    const float* __restrict__ x, float* __restrict__ outsum)
{
    __shared__ float red[8];
    const long long base = (long long)blockIdx.x * PLANE;
    float s = 0.f;
    for (int i = threadIdx.x; i < PLANE; i += 256) s += x[base + i];
    #pragma unroll
    for (int off = 16; off; off >>= 1) s += __shfl_xor(s, off, 32);
    if ((threadIdx.x & 31) == 0) red[threadIdx.x >> 5] = s;
    __syncthreads();
    if (threadIdx.x == 0) {
        float t = 0.f;
        #pragma unroll
        for (int i = 0; i < 8; ++i) t += red[i];
        outsum[blockIdx.x] = t;
    }
}

// ---------------------------------------------------------------------------
// Kernel 2: f = sigmoid( (sums*scale) @ W^T ), via V_WMMA_F32_16X16X4_F32.
//   sums : (16, 256) row-major   (A matrix, M=16, K=256)
//   W    : (56, 256) row-major   (B[k][o] = W[o][k], N=56 padded to 64)
//   fout : (16, 56)  row-major
// One block, 4 waves; wave w computes N-tile [16w, 16w+16).
// A layout per ISA 16x4 f32 table: lanes 0-15 -> K={k0,k0+1}, lanes 16-31 ->
// K={k0+2,k0+3}; B mirrored; D per 16x16 f32 C/D table.
// ---------------------------------------------------------------------------
__global__ __launch_bounds__(128) void gemm_sigmoid_wmma(
    const float* __restrict__ sums, const float* __restrict__ W,
    float* __restrict__ fout, float scale)
{
    const int wave = threadIdx.x >> 5;         // 0..3 : N tile
    const int lane = threadIdx.x & 31;
    const int m    = lane & 15;                // A row / B-D column-in-tile
    const int kh   = lane >> 4;                // 0 or 1 (K half)
    const int ncol = wave * 16 + m;            // output column 0..63
    const int ncl  = ncol < FDIM ? ncol : (FDIM - 1);  // clamp for safe load
    const bool nok = (ncol < FDIM);

    v8f acc = {};
    for (int k0 = 0; k0 < CH; k0 += 4) {
        const int ka = k0 + 2 * kh;
        v2f a, b;
        a.x = sums[m * CH + ka]     * scale;
        a.y = sums[m * CH + ka + 1] * scale;
        const float b0 = W[ncl * CH + ka];
        const float b1 = W[ncl * CH + ka + 1];
        b.x = nok ? b0 : 0.0f;                 // v_cndmask, EXEC stays all-1s
        b.y = nok ? b1 : 0.0f;
        acc = __builtin_amdgcn_wmma_f32_16x16x4_f32(
                  false, a, false, b, (short)0, acc, false, false);
    }
    if (nok) {
        #pragma unroll
        for (int r = 0; r < 8; ++r) {
            const int mr = r + 8 * kh;         // D: lanes16-31 hold M=r+8
            const float v = acc[r];
            fout[mr * FDIM + ncol] = 1.0f / (1.0f + expf(-v));
        }
    }
}

// reflect index for pad=3, extent 96 (numpy 'reflect', no edge repeat)
__device__ __forceinline__ int refl96(int j) {
    j = (j < 0) ? -j : j;
    return (j >= 96) ? (190 - j) : j;
}

// ---------------------------------------------------------------------------
// Kernel 3: horizontal 7-tap reflect conv, plane staged in LDS via async copy.
// Writes y1 and the per-plane sum (feeds second projection). One block/plane.
// ---------------------------------------------------------------------------
__global__ __launch_bounds__(256) void hconv_kernel(
    const float* __restrict__ x, const float* __restrict__ f,
    float* __restrict__ y, float* __restrict__ m2sum)
{
    __shared__ __align__(16) float tile[PLANE];
    __shared__ float red[8];
    const int nc = blockIdx.x;
    const int n  = nc >> 8;                    // / CH
    const int g  = (nc & (CH - 1)) >> 5;       // c / 32
    const long long base = (long long)nc * PLANE;

    async_plane_to_lds(x + base, tile);
    float wt[KTAP];
    #pragma unroll
    for (int i = 0; i < KTAP; ++i) wt[i] = f[n * FDIM + g * KTAP + i];
    __syncthreads();

    float s = 0.f;
    for (int idx = threadIdx.x; idx < PLANE; idx += 256) {
        const int h = idx / WW_, w = idx - h * WW_;
        const float* row = &tile[h * WW_];
        float v = 0.f;
        #pragma unroll
        for (int i = 0; i < KTAP; ++i) v += row[refl96(w + i - 3)] * wt[i];
        y[base + idx] = v;
        s += v;
    }
    #pragma unroll
    for (int off = 16; off; off >>= 1) s += __shfl_xor(s, off, 32);
    if ((threadIdx.x & 31) == 0) red[threadIdx.x >> 5] = s;
    __syncthreads();
    if (threadIdx.x == 0) {
        float t = 0.f;
        #pragma unroll
        for (int i = 0; i < 8; ++i) t += red[i];
        m2sum[nc] = t;                         // one block per plane: plain store
    }
}

// ---------------------------------------------------------------------------
// Kernel 4: vertical 7-tap reflect conv on y (block-local in-place via LDS)
// fused with final affine: out = gamma[c]*v + beta[c]*x.
// ---------------------------------------------------------------------------
__global__ __launch_bounds__(256) void vconv_combine_kernel(
    float* __restrict__ y, const float* __restrict__ f,
    const float* __restrict__ x,
    const float* __restrict__ gamma, const float* __restrict__ beta)
{
    __shared__ __align__(16) float tile[PLANE];
    const int nc = blockIdx.x;
    const int n  = nc >> 8;
    const int c  = nc & (CH - 1);
    const int g  = c >> 5;
    const long long base = (long long)nc * PLANE;

    async_plane_to_lds(y + base, tile);
    float wt[KTAP];
    #pragma unroll
    for (int i = 0; i < KTAP; ++i) wt[i] = f[n * FDIM + g * KTAP + i];
    const float ga = gamma[c], be = beta[c];
    __syncthreads();

    for (int idx = threadIdx.x; idx < PLANE; idx += 256) {
        const int h = idx / WW_, w = idx - h * WW_;
        float v = 0.f;
        #pragma unroll
        for (int i = 0; i < KTAP; ++i) v += tile[refl96(h + i - 3) * WW_ + w] * wt[i];
        y[base + idx] = ga * v + be * x[base + idx];
    }
}

// ---------------------------------------------------------------------------
extern "C" void kernel_launch(void* const* d_in, const int* in_sizes, int n_in,
                              void* d_out, int out_size, void* d_ws, size_t ws_size,
                              hipStream_t stream) {
    const float* x     = (const float*)d_in[0];
    const float* Wh    = (const float*)d_in[1];
    const float* Ww    = (const float*)d_in[2];
    const float* gamma = (const float*)d_in[3];
    const float* beta  = (const float*)d_in[4];
    float* out = (float*)d_out;
    float* ws  = (float*)d_ws;

    float* m1 = ws;                 // 4096 floats: sums of x planes
    float* m2 = ws + 4096;          // 4096 floats: sums of y1 planes
    float* fh = ws + 8192;          // 896 floats : sigmoid weights, h pass
    float* fv = ws + 8192 + 896;    // 896 floats : sigmoid weights, v pass

    const int nplanes = NB * CH;    // 4096
    const float scale = 1.0f / (float)PLANE;

    plane_sum_kernel    <<<nplanes, 256, 0, stream>>>(x, m1);
    gemm_sigmoid_wmma   <<<1, 128, 0, stream>>>(m1, Wh, fh, scale);
    hconv_kernel        <<<nplanes, 256, 0, stream>>>(x, fh, out, m2);
    gemm_sigmoid_wmma   <<<1, 128, 0, stream>>>(m2, Ww, fv, scale);
    vconv_combine_kernel<<<nplanes, 256, 0, stream>>>(out, fv, x, gamma, beta);
}